// YoloDetectionLayer_30588757082675
// MI455X (gfx1250) — compile-verified
//
#include <hip/hip_runtime.h>
#include <hip/hip_bf16.h>
#include <math.h>

// ---------------- problem constants (match reference) ----------------
#define NCLS   80
#define NLOC   3
#define BB     16           // batch
#define TT     50           // targets per image
#define GD     64           // grid (in_h == in_w == 64)
#define A0IDX  6            // scale_w = 512/64 = 8 -> anchors 6..8
#define CELLS  (BB * NLOC * GD * GD)   // 196,608
#define CHSTR  (GD * GD)               // channel stride in floats (4096)
#define EPSF   1e-12f

typedef float v2f __attribute__((ext_vector_type(2)));
typedef float v8f __attribute__((ext_vector_type(8)));

// anchors in grid units (ANCHORS / 8)
__device__ __constant__ float c_aw[9] = {1.25f, 2.0f, 4.125f, 3.75f, 7.75f, 7.375f, 14.5f, 19.5f, 46.625f};
__device__ __constant__ float c_ah[9] = {1.625f, 3.75f, 2.875f, 7.625f, 5.625f, 14.875f, 11.25f, 24.75f, 40.75f};

__device__ __forceinline__ float sigmoidf_(float z) { return 1.0f / (1.0f + expf(-z)); }

// Wave(32) sum through the matrix pipe: A = lane partials (16x4 f32), B = ones.
// D[m][n] = v_m + v_{m+16}; lane L holds D rows {0..7} (L<16) or {8..15} (L>=16)
// across its 8 D regs, so one 8-way add + one xor-16 permute completes the sum.
__device__ __forceinline__ float wave_sum32(float v) {
#if __has_builtin(__builtin_amdgcn_wmma_f32_16x16x4_f32)
    v2f a; a[0] = v;    a[1] = 0.0f;
    v2f b; b[0] = 1.0f; b[1] = 1.0f;
    v8f c = {};
    v8f d = __builtin_amdgcn_wmma_f32_16x16x4_f32(false, a, false, b, (short)0, c, false, false);
    float s = ((d[0] + d[1]) + (d[2] + d[3])) + ((d[4] + d[5]) + (d[6] + d[7]));
    s += __shfl_xor(s, 16, 32);
    return s;   // full wave sum, broadcast to all lanes
#else
    for (int m = 16; m >= 1; m >>= 1) v += __shfl_xor(v, m, 32);
    return v;
#endif
}

// ---------------- kernel 0: zero flags + accumulators ----------------
__global__ void yolo_init_kernel(int* __restrict__ ws, int n) {
    for (int i = blockIdx.x * blockDim.x + threadIdx.x; i < n; i += gridDim.x * blockDim.x)
        ws[i] = 0;
}

// ---------------- kernel 1: sparse target construction ----------------
// One thread per (b,t). Mirrors _get_target's scatters without dense grids.
__global__ void yolo_targets_kernel(const float* __restrict__ gt,
                                    int* __restrict__ zeroFlag,   // [CELLS] 1 = noobj zeroed
                                    int* __restrict__ maskTgt,    // [CELLS] 0 = none, else t+1
                                    float* __restrict__ tvals) {  // [BB*TT][8]
    int gid = blockIdx.x * blockDim.x + threadIdx.x;
    if (gid >= BB * TT) return;
    int b = gid / TT;
    int t = gid - b * TT;

    const float* g = gt + (size_t)gid * 5;
    float x = g[0], y = g[1], w = g[2], h = g[3], cf = g[4];
    bool valid = (x + y + w + h + cf) > 0.0f;

    float gx = x * GD, gy = y * GD, gw = w * GD, gh = h * GD;
    int gi = (int)gy;   // note: gi from y, gj from x (reference semantics)
    int gj = (int)gx;
    bool inb = (gi >= 0) && (gi < GD) && (gj >= 0) && (gj < GD);

    float best_iou = -1.0f; int best = 0;
    float lio[NLOC];
    #pragma unroll
    for (int k = 0; k < 9; ++k) {
        float inter = fminf(gw, c_aw[k]) * fminf(gh, c_ah[k]);
        float uni   = gw * gh + c_aw[k] * c_ah[k] - inter;
        float iou   = inter / (uni + 1e-16f);
        if (iou > best_iou) { best_iou = iou; best = k; }   // first-max wins (argmax)
        if (k >= A0IDX) lio[k - A0IDX] = iou;
    }

    if (valid && inb) {
        #pragma unroll
        for (int k = 0; k < NLOC; ++k)
            if (lio[k] > 0.5f)
                zeroFlag[((b * NLOC + k) * GD + gi) * GD + gj] = 1;  // benign write race (all write 1)
    }

    int local = best - A0IDX;
    if (valid && inb && local >= 0 && local < NLOC) {
        float* tv = tvals + (size_t)gid * 8;
        tv[0] = gx - (float)gj;
        tv[1] = gy - (float)gi;
        tv[2] = logf(gw / c_aw[best] + 1e-16f);
        tv[3] = logf(gh / c_ah[best] + 1e-16f);
        tv[4] = cf;                       // class index as float
        maskTgt[((b * NLOC + local) * GD + gi) * GD + gj] = t + 1;  // duplicate cells: one winner
    }
}

// ---------------- kernel 2: fused per-cell loss + WMMA wave reduction ----------------
// acc layout: 0:sx 1:sy 2:sw 3:sh 4:sobj 5:snoobj 6:scls 7:n_obj 8:n_noobj
__global__ __launch_bounds__(256) void yolo_main_kernel(const float* __restrict__ pred,
                                                        const int* __restrict__ zeroFlag,
                                                        const int* __restrict__ maskTgt,
                                                        const float* __restrict__ tvals,
                                                        float* __restrict__ acc) {
    int tid = blockIdx.x * 256 + threadIdx.x;   // grid sized exactly to CELLS
    int j = tid & (GD - 1);
    int i = (tid >> 6) & (GD - 1);
    int a = (tid >> 12) % NLOC;
    int b = tid / (NLOC * GD * GD);

    const float* base = pred + (((size_t)(b * (NLOC * (5 + NCLS)) + a * (5 + NCLS))) * GD + i) * GD + j;

    float partial[9];
    #pragma unroll
    for (int k = 0; k < 9; ++k) partial[k] = 0.0f;

    // conf channel: needed at every cell
    float pconf = sigmoidf_(base[4 * CHSTR]);
    pconf = fminf(fmaxf(pconf, EPSF), 1.0f - EPSF);
    int m = maskTgt[tid];
    float yconf = (m != 0) ? 1.0f : 0.0f;
    float bce_conf = -(yconf * logf(pconf) + (1.0f - yconf) * logf(1.0f - pconf));

    if (zeroFlag[tid] == 0) {       // noobj == 1 here
        partial[5] = bce_conf;
        partial[8] = 1.0f;
    }

    if (m != 0) {                   // masked cell: box + obj + class losses
        const float* tv = tvals + (size_t)(b * TT + (m - 1)) * 8;
        float dx = sigmoidf_(base[0 * CHSTR]) - tv[0];
        float dy = sigmoidf_(base[1 * CHSTR]) - tv[1];
        float dw = base[2 * CHSTR] - tv[2];
        float dh = base[3 * CHSTR] - tv[3];
        partial[0] = dx * dx;
        partial[1] = dy * dy;
        partial[2] = dw * dw;
        partial[3] = dh * dh;
        partial[4] = bce_conf;
        partial[7] = 1.0f;
        int cls = (int)tv[4];
        float scls = 0.0f;
        for (int c2 = 0; c2 < NCLS; ++c2) {
            float p = sigmoidf_(base[(5 + c2) * CHSTR]);
            p = fminf(fmaxf(p, EPSF), 1.0f - EPSF);
            float yc = (c2 == cls) ? 1.0f : 0.0f;
            scls += -(yc * logf(p) + (1.0f - yc) * logf(1.0f - p));
        }
        partial[6] = scls;
    }

    // Reduction: WMMA wave-sum -> LDS combine -> one global atomic per block.
    __shared__ float sacc[9];
    if (threadIdx.x < 9) sacc[threadIdx.x] = 0.0f;
    __syncthreads();

    #pragma unroll
    for (int k = 0; k < 9; ++k) {
        float s = wave_sum32(partial[k]);           // EXEC all-ones here (no divergence)
        if ((threadIdx.x & 31) == 0) atomicAdd(&sacc[k], s);
    }
    __syncthreads();
    if (threadIdx.x < 9) atomicAdd(&acc[threadIdx.x], sacc[threadIdx.x]);
}

// ---------------- kernel 3: finalize ----------------
__global__ void yolo_finalize_kernel(const float* __restrict__ acc, float* __restrict__ out) {
    float n_obj   = fmaxf(acc[7], 1.0f);
    float n_noobj = fmaxf(acc[8], 1.0f);
    float bbox = 5.0f * (acc[0] + acc[1] + acc[2] + acc[3]) / n_obj;          // LAMBDA_COORD
    float obj  = acc[4] / n_obj + 0.5f * acc[5] / n_noobj;                    // OBJ + NOOBJ
    float cls  = acc[6] / (n_obj * (float)NCLS);                              // LAMBDA_CLS
    out[0] = bbox + obj + cls;
    out[1] = bbox;
    out[2] = obj;
    out[3] = cls;
}

extern "C" void kernel_launch(void* const* d_in, const int* in_sizes, int n_in,
                              void* d_out, int out_size, void* d_ws, size_t ws_size,
                              hipStream_t stream) {
    const float* backbone = (const float*)d_in[0];   // (16,255,64,64) fp32
    const float* gt       = (const float*)d_in[1];   // (16,50,5) fp32
    float* out = (float*)d_out;                      // 4 scalars

    // workspace layout
    char* ws = (char*)d_ws;
    float* acc     = (float*)ws;                          // 16 floats (use 9)
    int*   zeroFl  = (int*)(ws + 64);                     // CELLS ints
    int*   maskTg  = (int*)(ws + 64 + 4 * (size_t)CELLS); // CELLS ints
    float* tvals   = (float*)(ws + 64 + 8 * (size_t)CELLS); // BB*TT*8 floats
    (void)in_sizes; (void)n_in; (void)out_size; (void)ws_size;

    int initWords = 16 + 2 * CELLS;
    yolo_init_kernel<<<(initWords + 255) / 256, 256, 0, stream>>>((int*)ws, initWords);
    yolo_targets_kernel<<<(BB * TT + 255) / 256, 256, 0, stream>>>(gt, zeroFl, maskTg, tvals);
    yolo_main_kernel<<<CELLS / 256, 256, 0, stream>>>(backbone, zeroFl, maskTg, tvals, acc);
    yolo_finalize_kernel<<<1, 1, 0, stream>>>(acc, out);
}